// Deform_11209864642861
// MI455X (gfx1250) — compile-verified
//
#include <hip/hip_runtime.h>

// Bilinear grid_sample, faithful to the JAX/TF reference:
//   source : (1, 256, 256, 3) f32   -- single image, batch-gather collapses to it
//   motions: (8, 11, 256, 256, 2) f32, coords in [-1,1)
//   out    : (88, 65536, 3) f32
//
// Memory-bound (~115 MB HBM traffic -> ~5us floor @ 23.3 TB/s). No matrix ops
// apply (data-dependent gathers). CDNA5-specific tuning: NT temporal hints on
// the streamed motion loads / output stores (keep the 768 KB source hot in the
// 192 MB L2 across its 88x reuse), gfx1250 global_prefetch on the motion
// stream, and coalesced b64/b96 access patterns under wave32.

#define SRC_H 256
#define SRC_W 256
#define SRC_C 3

typedef float v2f __attribute__((ext_vector_type(2)));

__global__ __launch_bounds__(256) void deform_bilinear_kernel(
    const float* __restrict__ src,   // (256,256,3)
    const float* __restrict__ mot,   // (total,2) flattened (n,y,x,2)
    float* __restrict__ out,         // (total,3)
    int total)
{
    int idx = blockIdx.x * blockDim.x + threadIdx.x;
    if (idx >= total) return;

    // One lane per wave prefetches the motion stream ~64KB ahead (speculative;
    // out-of-range prefetches are dropped silently per ISA 10.5).
    if ((threadIdx.x & 31) == 0) {
        __builtin_prefetch(mot + 2u * (unsigned)idx + 16384u, 0, 1);
    }

    // Streamed exactly once: non-temporal load (TH=NT) so it doesn't evict the
    // L2-resident source image.
    v2f g = __builtin_nontemporal_load((const v2f*)mot + idx);

    // Normalized -> pixel coords (reference uses grid's H/W == source H/W here)
    float x = (g.x + 1.0f) * (SRC_W * 0.5f) - 0.5f;
    float y = (g.y + 1.0f) * (SRC_H * 0.5f) - 0.5f;

    float xw = floorf(x);
    float yn = floorf(y);
    float xe = xw + 1.0f;
    float ys = yn + 1.0f;

    float fw = x - xw;          // frac toward east
    float fe = 1.0f - fw;
    float fn = y - yn;          // frac toward south
    float fs = 1.0f - fn;

    // corner weights (names follow reference)
    float w_nw = fs * fe;
    float w_ne = fs * fw;
    float w_sw = fn * fe;
    float w_se = fn * fw;

    // in-bounds masks, float-compare semantics identical to reference inb()
    float mW = (xw > -1.0f && xw < (float)SRC_W) ? 1.0f : 0.0f;
    float mN = (yn > -1.0f && yn < (float)SRC_H) ? 1.0f : 0.0f;
    float mE = (xe > -1.0f && xe < (float)SRC_W) ? 1.0f : 0.0f;
    float mS = (ys > -1.0f && ys < (float)SRC_H) ? 1.0f : 0.0f;

    float m_nw = mN * mW;
    float m_ne = mN * mE;
    float m_sw = mS * mW;
    float m_se = mS * mE;

    float r = 0.0f, gg = 0.0f, b = 0.0f;

    // Branch-free masked gather: mask multiplies the float coord before the
    // int cast (OOB -> index 0) and the gathered value (OOB -> contributes 0),
    // exactly like the reference. 3 contiguous floats per corner -> b96-able.
    #define CORNER(M, YY, XX, WGT)                                  \
    {                                                               \
        int yi = (int)((M) * (YY));                                 \
        int xi = (int)((M) * (XX));                                 \
        const float* p = src + (yi * SRC_W + xi) * SRC_C;           \
        float cw = (WGT) * (M);                                     \
        r  += cw * p[0];                                            \
        gg += cw * p[1];                                            \
        b  += cw * p[2];                                            \
    }

    CORNER(m_nw, yn, xw, w_nw)
    CORNER(m_sw, ys, xw, w_sw)
    CORNER(m_ne, yn, xe, w_ne)
    CORNER(m_se, ys, xe, w_se)

    #undef CORNER

    // Written once, never re-read: non-temporal stores (TH=NT), 12B/lane
    // contiguous across the wave -> fully coalesced (mergeable to b96).
    float* o = out + (size_t)idx * 3u;
    __builtin_nontemporal_store(r,  o + 0);
    __builtin_nontemporal_store(gg, o + 1);
    __builtin_nontemporal_store(b,  o + 2);
}

extern "C" void kernel_launch(void* const* d_in, const int* in_sizes, int n_in,
                              void* d_out, int out_size, void* d_ws, size_t ws_size,
                              hipStream_t stream) {
    const float* src = (const float*)d_in[0];   // (1,256,256,3) f32
    const float* mot = (const float*)d_in[1];   // (8,11,256,256,2) f32

    float* out = (float*)d_out;                 // (88,65536,3) f32

    const int total = in_sizes[1] / 2;          // number of sampled pixels
    const int block = 256;                      // 8 wave32s
    const int grid  = (total + block - 1) / block;

    deform_bilinear_kernel<<<grid, block, 0, stream>>>(src, mot, out, total);
}